// Graph2VecSortPooling_40948218200429
// MI455X (gfx1250) — compile-verified
//
#include <hip/hip_runtime.h>
#include <hip/hip_bf16.h>
#include <float.h>

// ---------------- problem constants (match reference) ----------------
#define N_NODES   100352
#define NUM_G     512
#define NPG       196
#define N_EDGES   1605632
#define IN_C      128
#define HID_C     64
#define OUT_C     32
#define TOPK      30

typedef __attribute__((ext_vector_type(2))) float v2f;
typedef __attribute__((ext_vector_type(8))) float v8f;

// ---------------- degree / normalization ----------------
__global__ void init_deg_kernel(float* deg) {
    int i = blockIdx.x * blockDim.x + threadIdx.x;
    if (i < N_NODES) deg[i] = 1.0f;           // +1 self loop
}

__global__ void deg_scatter_kernel(const int* __restrict__ ei, float* deg) {
    int e = blockIdx.x * blockDim.x + threadIdx.x;
    if (e < N_EDGES) {
        int dst = ei[N_EDGES + e];
        atomicAdd(&deg[dst], 1.0f);
    }
}

__global__ void rsqrt_kernel(const float* __restrict__ deg, float* dis) {
    int i = blockIdx.x * blockDim.x + threadIdx.x;
    if (i < N_NODES) dis[i] = rsqrtf(deg[i]);
}

// ---------------- WMMA fp32 GEMM: Y[M,N] = X[M,K] @ W[K,N] ----------------
// One 256-thread block = 8 waves; each wave owns one 16-row M tile and the
// full N strip (N/16 accumulators). W staged in LDS. M must be % 128 == 0.
template<int K, int N>
__global__ __launch_bounds__(256) void gemm_wmma_f32(const float* __restrict__ X,
                                                     const float* __restrict__ W,
                                                     float* __restrict__ Y) {
    constexpr int NT = N / 16;
    __shared__ float wl[K * N];
    for (int i = threadIdx.x; i < K * N; i += 256) wl[i] = W[i];
    __syncthreads();

    const int wave = threadIdx.x >> 5;
    const int lane = threadIdx.x & 31;
    const int row0 = (blockIdx.x * 8 + wave) * 16;
    const int r  = lane & 15;     // row-in-tile (A), col-in-tile (B/D)
    const int kh = lane >> 4;     // which K pair (0 -> K 0,1 ; 1 -> K 2,3)

    v8f acc[NT];
#pragma unroll
    for (int t = 0; t < NT; ++t) acc[t] = (v8f){};

    const float* xr = X + (size_t)(row0 + r) * K + 2 * kh;
#pragma unroll 4
    for (int kb = 0; kb < K; kb += 4) {
        v2f a;
        a.x = xr[kb];
        a.y = xr[kb + 1];
#pragma unroll
        for (int t = 0; t < NT; ++t) {
            v2f b;
            b.x = wl[(kb + 2 * kh) * N + t * 16 + r];
            b.y = wl[(kb + 2 * kh + 1) * N + t * 16 + r];
            // D = A(16x4) * B(4x16) + C, exact fp32
            acc[t] = __builtin_amdgcn_wmma_f32_16x16x4_f32(
                false, a, false, b, (short)0, acc[t], false, false);
        }
    }

#pragma unroll
    for (int t = 0; t < NT; ++t)
#pragma unroll
        for (int v = 0; v < 8; ++v)
            Y[(size_t)(row0 + v + 8 * kh) * N + t * 16 + r] = acc[t][v];
}

// ---------------- self-loop term: h = dis^2 * xw (full write) ----------------
template<int C>
__global__ void self_term_kernel(const float* __restrict__ xw,
                                 const float* __restrict__ dis,
                                 float* __restrict__ h) {
    long idx = (long)blockIdx.x * blockDim.x + threadIdx.x;
    const long total = (long)N_NODES * C;
    if (idx < total) {
        int node = (int)(idx / C);
        float d = dis[node];
        h[idx] = d * d * xw[idx];
    }
}

// ---------------- edge scatter: h[dst] += dis[src]*dis[dst]*xw[src] ----------------
// one thread per (edge, 4-channel group): float4 gather + 4 atomics (L2-resident)
template<int C>
__global__ void edge_scatter_kernel(const int* __restrict__ ei,
                                    const float* __restrict__ xw,
                                    const float* __restrict__ dis,
                                    float* __restrict__ h) {
    constexpr int G4 = C / 4;
    long idx = (long)blockIdx.x * blockDim.x + threadIdx.x;
    const long total = (long)N_EDGES * G4;
    if (idx >= total) return;
    int e  = (int)(idx / G4);
    int cg = (int)(idx % G4);
    int src = ei[e];
    int dst = ei[N_EDGES + e];
    float w = dis[src] * dis[dst];
    const float4 v = *(const float4*)(xw + (size_t)src * C + cg * 4);
    float* hp = h + (size_t)dst * C + cg * 4;
    atomicAdd(hp + 0, w * v.x);
    atomicAdd(hp + 1, w * v.y);
    atomicAdd(hp + 2, w * v.z);
    atomicAdd(hp + 3, w * v.w);
}

// ---------------- bias (+ optional relu) ----------------
template<int C, bool RELU>
__global__ void bias_act_kernel(float* __restrict__ h, const float* __restrict__ b) {
    long idx = (long)blockIdx.x * blockDim.x + threadIdx.x;
    const long total = (long)N_NODES * C;
    if (idx < total) {
        int c = (int)(idx % C);
        float v = h[idx] + b[c];
        h[idx] = RELU ? fmaxf(v, 0.0f) : v;
    }
}

// ---------------- sort pooling: per-graph top-30 by last channel ----------------
__global__ __launch_bounds__(256) void sortpool_kernel(const float* __restrict__ h2,
                                                       float* __restrict__ out) {
    int g = blockIdx.x;
    int tid = threadIdx.x;
    __shared__ float key[NPG];
    __shared__ float rkey[256];
    __shared__ int   ridx[256];
    __shared__ int   order[TOPK];

    for (int j = tid; j < NPG; j += 256)
        key[j] = h2[(size_t)(g * NPG + j) * OUT_C + (OUT_C - 1)];
    __syncthreads();

    for (int it = 0; it < TOPK; ++it) {
        float bk = -FLT_MAX;
        int   bi = -1;
        for (int j = tid; j < NPG; j += 256) {
            float k = key[j];
            if (k > bk) { bk = k; bi = j; }     // strict '>' keeps smallest index
        }
        rkey[tid] = bk; ridx[tid] = bi;
        __syncthreads();
        for (int s = 128; s > 0; s >>= 1) {
            if (tid < s) {
                float ok = rkey[tid + s]; int oi = ridx[tid + s];
                bool take = (ok > rkey[tid]) ||
                            (ok == rkey[tid] && oi != -1 &&
                             (ridx[tid] == -1 || oi < ridx[tid]));
                if (take) { rkey[tid] = ok; ridx[tid] = oi; }
            }
            __syncthreads();
        }
        if (tid == 0) {
            int best = ridx[0];
            order[it] = best;
            key[best] = -FLT_MAX;               // remove from next rounds
        }
        __syncthreads();
    }

    // gather top-30 rows (graphs are exactly full: no fill padding possible)
    for (int p = tid; p < TOPK * OUT_C; p += 256) {
        int rrow = p >> 5;           // /32
        int c    = p & (OUT_C - 1);
        out[(size_t)g * (TOPK * OUT_C) + p] =
            h2[(size_t)(g * NPG + order[rrow]) * OUT_C + c];
    }
}

// ---------------- launch ----------------
extern "C" void kernel_launch(void* const* d_in, const int* in_sizes, int n_in,
                              void* d_out, int out_size, void* d_ws, size_t ws_size,
                              hipStream_t stream) {
    const float* x  = (const float*)d_in[0];
    const int*   ei = (const int*)d_in[1];
    // d_in[2] = batch (implicit: equal graphs of 196 nodes)
    const float* W1 = (const float*)d_in[3];
    const float* b1 = (const float*)d_in[4];
    const float* W2 = (const float*)d_in[5];
    const float* b2 = (const float*)d_in[6];
    float* out = (float*)d_out;

    // workspace carve-up (bytes), 256-aligned sections
    char* ws = (char*)d_ws;
    size_t off = 0;
    auto carve = [&](size_t bytes) { char* p = ws + off; off = (off + bytes + 255) & ~(size_t)255; return p; };
    float* deg = (float*)carve((size_t)N_NODES * 4);
    float* dis = (float*)carve((size_t)N_NODES * 4);
    float* xw1 = (float*)carve((size_t)N_NODES * HID_C * 4);
    float* h1  = (float*)carve((size_t)N_NODES * HID_C * 4);
    float* xw2 = (float*)carve((size_t)N_NODES * OUT_C * 4);
    float* h2  = (float*)carve((size_t)N_NODES * OUT_C * 4);
    (void)ws_size; (void)in_sizes; (void)n_in; (void)out_size;

    const int T = 256;
    auto blk = [](long n, int t) { return (int)((n + t - 1) / t); };

    // degree + normalization
    init_deg_kernel<<<blk(N_NODES, T), T, 0, stream>>>(deg);
    deg_scatter_kernel<<<blk(N_EDGES, T), T, 0, stream>>>(ei, deg);
    rsqrt_kernel<<<blk(N_NODES, T), T, 0, stream>>>(deg, dis);

    // ---- layer 1: h1 = relu(GCN(x, W1, b1)) ----
    gemm_wmma_f32<IN_C, HID_C><<<N_NODES / 128, T, 0, stream>>>(x, W1, xw1);
    self_term_kernel<HID_C><<<blk((long)N_NODES * HID_C, T), T, 0, stream>>>(xw1, dis, h1);
    edge_scatter_kernel<HID_C><<<blk((long)N_EDGES * (HID_C / 4), T), T, 0, stream>>>(ei, xw1, dis, h1);
    bias_act_kernel<HID_C, true><<<blk((long)N_NODES * HID_C, T), T, 0, stream>>>(h1, b1);

    // ---- layer 2: h2 = GCN(h1, W2, b2) ----
    gemm_wmma_f32<HID_C, OUT_C><<<N_NODES / 128, T, 0, stream>>>(h1, W2, xw2);
    self_term_kernel<OUT_C><<<blk((long)N_NODES * OUT_C, T), T, 0, stream>>>(xw2, dis, h2);
    edge_scatter_kernel<OUT_C><<<blk((long)N_EDGES * (OUT_C / 4), T), T, 0, stream>>>(ei, xw2, dis, h2);
    bias_act_kernel<OUT_C, false><<<blk((long)N_NODES * OUT_C, T), T, 0, stream>>>(h2, b2);

    // ---- sort pooling ----
    sortpool_kernel<<<NUM_G, T, 0, stream>>>(h2, out);
}